// KPFBNC_8770323218599
// MI455X (gfx1250) — compile-verified
//
#include <hip/hip_runtime.h>
#include <math.h>

typedef __attribute__((ext_vector_type(2))) float v2f;
typedef __attribute__((ext_vector_type(4))) float v4f;
typedef __attribute__((ext_vector_type(8))) float v8f;
typedef __attribute__((ext_vector_type(4))) unsigned u32x4;
typedef __attribute__((ext_vector_type(8))) unsigned u32x8;

#define B_ 64
#define N_ 200
#define P_ 400
#define T_ 1200
#define STRIDE_ 25
#define EPS_ 1e-9f

// workspace layout (in floats)
#define WS_NP    0                      // normalized_patterns P*N
#define WS_PES   (WS_NP + P_*N_)        // B*P
#define WS_TOPV  (WS_PES + B_*P_)       // P*3 top values
#define WS_TOPI  (WS_TOPV + P_*3)       // P*3 top indices (as int)
#define WS_NORM  (WS_TOPI + P_*3)       // scalar accumulator
#define WS_ORTH  (WS_NORM + 1)          // scalar accumulator

// ---- Tensor Data Mover: 2D tile load, global -> LDS (D# per CDNA5 ISA) --
// dim0 = contiguous direction (data_size=4B units), dim1 rows with stride0.
__device__ __forceinline__ void tdm_load_2d(unsigned lds_addr, const void* gaddr,
                                            unsigned tile_d0, unsigned tile_d1,
                                            unsigned tensor_d0, unsigned tensor_d1,
                                            unsigned long long stride0) {
    unsigned long long ga = (unsigned long long)gaddr;
    u32x4 g0;
    g0[0] = 1u;                                   // count=1, user descriptor
    g0[1] = lds_addr;                             // LDS byte address
    g0[2] = (unsigned)ga;                         // global_addr[31:0]
    g0[3] = ((unsigned)(ga >> 32) & 0x01FFFFFFu)  // global_addr[56:32]
          | 0x80000000u;                          // type=2 ("image")
    u32x8 g1;
    g1[0] = 2u << 16;                             // workgroup_mask=0, data_size=4B
    g1[1] = (tensor_d0 & 0xFFFFu) << 16;          // tensor_dim0[15:0]
    g1[2] = ((tensor_d0 >> 16) & 0xFFFFu) | ((tensor_d1 & 0xFFFFu) << 16);
    g1[3] = ((tensor_d1 >> 16) & 0xFFFFu) | ((tile_d0 & 0xFFFFu) << 16);
    g1[4] = tile_d1 & 0xFFFFu;                    // tile_dim1 ; tile_dim2=0
    g1[5] = (unsigned)stride0;                    // tensor_dim0_stride[31:0]
    g1[6] = (unsigned)((stride0 >> 32) & 0xFFFFu);
    g1[7] = 0u;
    u32x4 gz = (u32x4)(0u);                       // groups 2/3 (dims unused)
    asm volatile("tensor_load_to_lds %0, %1, %2, %3"
                 :: "s"(g0), "s"(g1), "s"(gz), "s"(gz)
                 : "memory");
}

// ---------------- init: zero graphs region of d_out + scalar accumulators ----
__global__ __launch_bounds__(256) void kpf_init(float* __restrict__ out,
                                                float* __restrict__ ws) {
    unsigned i = blockIdx.x * 256u + threadIdx.x;
    if (i < (unsigned)(B_ * N_ * N_)) out[i] = 0.0f;
    if (i == 0) { ws[WS_NORM] = 0.0f; ws[WS_ORTH] = 0.0f; }
}

// ---------------- patterns: normalize rows, top-3 |.|, norm loss ------------
__global__ __launch_bounds__(64) void kpf_pat(const float* __restrict__ pat,
                                              float* __restrict__ ws) {
    int p = blockIdx.x;
    int tid = threadIdx.x;
    __shared__ float row[N_];
    __shared__ float red[64];
    const float* pr = pat + p * N_;
    float s = 0.0f;
    for (int n = tid; n < N_; n += 64) { float v = pr[n]; s += v * v; }
    red[tid] = s; __syncthreads();
    for (int o = 32; o > 0; o >>= 1) {
        if (tid < o) red[tid] += red[tid + o];
        __syncthreads();
    }
    float ssq = red[0];
    float inv = rsqrtf(ssq + EPS_);
    float* np = ws + WS_NP + p * N_;
    for (int n = tid; n < N_; n += 64) { float v = pr[n] * inv; np[n] = v; row[n] = v; }
    __syncthreads();
    if (tid == 0) {
        int i1 = 0, i2 = 0, i3 = 0;
        float m1 = -1.0f, m2 = -1.0f, m3 = -1.0f;
        for (int n = 0; n < N_; n++) {
            float a = fabsf(row[n]);
            if (a > m1)      { m3 = m2; i3 = i2; m2 = m1; i2 = i1; m1 = a; i1 = n; }
            else if (a > m2) { m3 = m2; i3 = i2; m2 = a; i2 = n; }
            else if (a > m3) { m3 = a; i3 = n; }
        }
        float v1 = row[i1], v2 = row[i2], v3 = row[i3];
        ws[WS_TOPV + p * 3 + 0] = v1;
        ws[WS_TOPV + p * 3 + 1] = v2;
        ws[WS_TOPV + p * 3 + 2] = v3;
        int* ti = (int*)(ws + WS_TOPI);
        ti[p * 3 + 0] = i1; ti[p * 3 + 1] = i2; ti[p * 3 + 2] = i3;
        float lentopp = sqrtf(v1 * v1 + v2 * v2 + v3 * v3);
        float lenp = sqrtf(ssq) * inv;           // ||normalized row||
        float d = 1.0f - lentopp / lenp;
        atomicAdd(ws + WS_NORM, d * d);
    }
}

// ---------------- fused sim GEMM (fp32 WMMA, TDM-fed) + pooling -> PES ------
// grid (25 p-tiles, 64 batches), 256 threads = 8 waves.
// T in 3 chunks of 400 = 25 WMMA row-tiles = 16 pool windows per chunk.
// K = 200 in 25 tiles of 8; x/B tiles double-buffered and DMA'd by the TDM.
#define KT_ 8
#define NKT_ 25
#define XBUF_ (KT_ * 400)      // 3200 floats
__global__ __launch_bounds__(256) void kpf_sim(const float* __restrict__ x,
                                               const float* __restrict__ np,
                                               const int*   __restrict__ length,
                                               float*       __restrict__ pes) {
    __shared__ float lds_x[2][XBUF_];      // 25.6 KB (double buffer)
    __shared__ float lds_b[2][16 * KT_];   //  1.0 KB ([p][k] from TDM)
    __shared__ float lds_sim[400 * 16];    // 25.6 KB
    __shared__ float lds_inv[400];
    __shared__ float lds_red[256];

    const int p0  = blockIdx.x * 16;
    const int b   = blockIdx.y;
    const int tid = threadIdx.x;
    const int wave_u = (__builtin_amdgcn_readfirstlane(tid) >> 5) & 7; // scalar
    const int lane = tid & 31;
    const int m  = lane & 15;          // A row / B,C,D column within tile
    const int hi = lane >> 4;          // lane-half selects K/M offset
    const float* xb = x + (size_t)b * (N_ * T_);
    const float* npp = np + (size_t)p0 * N_;

    // prologue: TDM the first x/B K-tile pair into buffer 0
    if (wave_u == 0) {
        tdm_load_2d((unsigned)(uintptr_t)&lds_x[0][0], xb,  400, KT_, 400, KT_, T_);
        tdm_load_2d((unsigned)(uintptr_t)&lds_b[0][0], npp, KT_, 16, KT_, 16, N_);
    }

    float accp = 0.0f, accn = 0.0f;
    for (int c = 0; c < 3; c++) {
        v8f acc[4];
        #pragma unroll
        for (int li = 0; li < 4; li++) acc[li] = (v8f){0,0,0,0,0,0,0,0};
        v4f fs4 = (v4f){0,0,0,0};

        for (int kt = 0; kt < NKT_; kt++) {
            const int s   = c * NKT_ + kt;
            const int buf = s & 1;
            const float* xbuf = &lds_x[buf][0];
            if (wave_u == 0) {
                if (s < 3 * NKT_ - 1) {             // prefetch tile s+1
                    int sn = s + 1;
                    int tc = sn / NKT_, kn = sn - tc * NKT_;
                    tdm_load_2d((unsigned)(uintptr_t)&lds_x[sn & 1][0],
                                xb + (size_t)(kn * KT_) * T_ + tc * 400,
                                400, KT_, 400, KT_, T_);
                    tdm_load_2d((unsigned)(uintptr_t)&lds_b[sn & 1][0],
                                npp + kn * KT_, KT_, 16, KT_, 16, N_);
                    __builtin_amdgcn_s_wait_tensorcnt(2);  // current pair done
                } else {
                    __builtin_amdgcn_s_wait_tensorcnt(0);
                }
            }
            __syncthreads();
            // B operands for this K-tile: registers, reused across 4 t-tiles
            v2f bv[2];
            #pragma unroll
            for (int k4 = 0; k4 < 2; k4++)
                bv[k4] = *(const v2f*)&lds_b[buf][m * KT_ + k4 * 4 + hi * 2];
            #pragma unroll
            for (int li = 0; li < 4; li++) {
                int tt = wave_u + li * 8;           // scalar; li<3 folds away
                if (tt < 25) {
                    v8f cacc = acc[li];
                    #pragma unroll
                    for (int k4 = 0; k4 < 2; k4++) {
                        int kl = k4 * 4 + hi * 2;
                        v2f av;
                        av.x = xbuf[kl * 400 + tt * 16 + m];
                        av.y = xbuf[(kl + 1) * 400 + tt * 16 + m];
                        cacc = __builtin_amdgcn_wmma_f32_16x16x4_f32(
                            false, av, false, bv[k4], (short)0, cacc, false, false);
                    }
                    acc[li] = cacc;
                }
            }
            // frame sum-of-squares: 100 threads own 4 t-columns each (float4)
            if (tid < 100) {
                #pragma unroll
                for (int kk = 0; kk < KT_; kk++) {
                    v4f v = *(const v4f*)&xbuf[kk * 400 + tid * 4];
                    fs4 += v * v;
                }
            }
            __syncthreads();
        }
        // frame inverse norms for this chunk
        if (tid < 100) {
            lds_inv[tid * 4 + 0] = rsqrtf(fs4.x + EPS_);
            lds_inv[tid * 4 + 1] = rsqrtf(fs4.y + EPS_);
            lds_inv[tid * 4 + 2] = rsqrtf(fs4.z + EPS_);
            lds_inv[tid * 4 + 3] = rsqrtf(fs4.w + EPS_);
        }
        __syncthreads();
        // scale by frame inv-norm, dump sim chunk
        #pragma unroll
        for (int li = 0; li < 4; li++) {
            int tt = wave_u + li * 8;
            if (tt < 25) {
                #pragma unroll
                for (int r = 0; r < 8; r++) {
                    int tl = tt * 16 + r + hi * 8;
                    lds_sim[tl * 16 + m] = acc[li][r] * lds_inv[tl];
                }
            }
        }
        __syncthreads();
        // pooling: thread = (window-in-chunk, p); 25-wide +/- max
        {
            int win = tid >> 4, pp = tid & 15;
            float pmax = -3.4e38f, nmax = -3.4e38f;
            int tb = win * STRIDE_;
            for (int j = 0; j < STRIDE_; j++) {
                float v = lds_sim[(tb + j) * 16 + pp];
                pmax = fmaxf(pmax, v);
                nmax = fmaxf(nmax, -v);
            }
            accp += pmax; accn += nmax;
        }
        __syncthreads();
    }
    lds_red[tid] = accp + accn;
    __syncthreads();
    if (tid < 16) {
        float tot = 0.0f;
        for (int w = 0; w < 16; w++) tot += lds_red[w * 16 + tid];
        float sap = floorf((float)length[b] / (float)STRIDE_);
        pes[b * P_ + p0 + tid] = tot / (2.0f * sap);
    }
}

// ---------------- orth loss: simpp = np @ np^T via fp32 WMMA ----------------
__global__ __launch_bounds__(32) void kpf_orth(const float* __restrict__ np,
                                               float* __restrict__ orth_acc) {
    const int pi0 = blockIdx.y * 16;
    const int pj0 = blockIdx.x * 16;
    const int lane = threadIdx.x & 31;
    const int m  = lane & 15;
    const int hi = lane >> 4;
    v8f c = (v8f){0,0,0,0,0,0,0,0};
    for (int k0 = 0; k0 < N_; k0 += 4) {
        int ka = k0 + hi * 2;
        v2f av, bv;
        av.x = np[(pi0 + m) * N_ + ka];
        av.y = np[(pi0 + m) * N_ + ka + 1];
        bv.x = np[(pj0 + m) * N_ + ka];
        bv.y = np[(pj0 + m) * N_ + ka + 1];
        c = __builtin_amdgcn_wmma_f32_16x16x4_f32(
            false, av, false, bv, (short)0, c, false, false);
    }
    float contrib = 0.0f;
    #pragma unroll
    for (int r = 0; r < 8; r++) {
        int i = pi0 + r + hi * 8;
        int j = pj0 + m;
        if (i != j) {
            float v = fabsf(c[r]);
            float rel = fmaxf(0.0f, (v - 0.3f) / (1.0f - 0.3f + 1e-5f));
            contrib += rel * rel;
        }
    }
    atomicAdd(orth_acc, contrib);
}

// ---------------- graphs: 3x3 sparse scatter per (b,p) ---------------------
__global__ __launch_bounds__(256) void kpf_graphs(const float* __restrict__ pes,
                                                  const float* __restrict__ topv,
                                                  const int*   __restrict__ topi,
                                                  float*       __restrict__ out) {
    int idx = blockIdx.x * 256 + threadIdx.x;
    if (idx >= B_ * P_) return;
    int b = idx / P_, p = idx - b * P_;
    float w = pes[idx];
    float v[3]; int ii[3];
    #pragma unroll
    for (int k = 0; k < 3; k++) { v[k] = topv[p * 3 + k]; ii[k] = topi[p * 3 + k]; }
    float* ob = out + (size_t)b * N_ * N_;
    #pragma unroll
    for (int a = 0; a < 3; a++)
        #pragma unroll
        for (int c2 = 0; c2 < 3; c2++) {
            int n = ii[a], mm = ii[c2];
            float val = w * v[a] * v[c2] * (n == mm ? 1.0f : 6.0f); // c = 3*400/200 = 6
            atomicAdd(&ob[n * N_ + mm], val);
        }
}

// ---------------- finalize scalars ------------------------------------------
__global__ void kpf_fin(const float* __restrict__ ws, float* __restrict__ out) {
    if (blockIdx.x == 0 && threadIdx.x == 0) {
        out[(size_t)B_ * N_ * N_ + 0] = ws[WS_NORM] / (float)P_;
        out[(size_t)B_ * N_ * N_ + 1] = ws[WS_ORTH] / (float)(P_ * P_);
    }
}

extern "C" void kernel_launch(void* const* d_in, const int* in_sizes, int n_in,
                              void* d_out, int out_size, void* d_ws, size_t ws_size,
                              hipStream_t stream) {
    const float* x        = (const float*)d_in[0];   // (B,N,T)
    const float* patterns = (const float*)d_in[1];   // (P,N)
    const int*   length   = (const int*)d_in[2];     // (B,)
    float* out = (float*)d_out;
    float* ws  = (float*)d_ws;

    float* np   = ws + WS_NP;
    float* pes  = ws + WS_PES;
    float* topv = ws + WS_TOPV;
    int*   topi = (int*)(ws + WS_TOPI);

    kpf_init<<<(B_ * N_ * N_ + 255) / 256, 256, 0, stream>>>(out, ws);
    kpf_pat<<<P_, 64, 0, stream>>>(patterns, ws);
    kpf_sim<<<dim3(P_ / 16, B_), 256, 0, stream>>>(x, np, length, pes);
    kpf_orth<<<dim3(P_ / 16, P_ / 16), 32, 0, stream>>>(np, ws + WS_ORTH);
    kpf_graphs<<<(B_ * P_ + 255) / 256, 256, 0, stream>>>(pes, topv, topi, out);
    kpf_fin<<<1, 1, 0, stream>>>(ws, out);
}